// TransformerDownSampling_32710470926926
// MI455X (gfx1250) — compile-verified
//
#include <hip/hip_runtime.h>
#include <cstdint>

// Problem constants (from the reference: B=16, N=8192, C=256, num_points=2048)
#define BATCH 16
#define NPTS  8192
#define NCH   256
#define MOUT  2048
#define FPS_T 1024
#define PPT   (NPTS / FPS_T)   // 8 points per thread

typedef unsigned int u32x4 __attribute__((ext_vector_type(4)));
typedef int          i32x8 __attribute__((ext_vector_type(8)));
typedef int          i32x4 __attribute__((ext_vector_type(4)));

#if defined(__has_builtin)
#if __has_builtin(__builtin_amdgcn_tensor_load_to_lds) && __has_builtin(__builtin_amdgcn_s_wait_tensorcnt)
#define USE_TDM 1
#endif
#endif

// wave32 argmax reduce; ties -> smallest index (matches jnp.argmax "first")
__device__ __forceinline__ void warpArgMax(float& d, int& n) {
#pragma unroll
  for (int off = 16; off > 0; off >>= 1) {
    float od = __shfl_xor(d, off, 32);
    int   on = __shfl_xor(n, off, 32);
    if (od > d || (od == d && on < n)) { d = od; n = on; }
  }
}

__device__ __forceinline__ float blockSum(float v, float* sbuf, int tid) {
#pragma unroll
  for (int off = 16; off > 0; off >>= 1) v += __shfl_xor(v, off, 32);
  if ((tid & 31) == 0) sbuf[tid >> 5] = v;
  __syncthreads();
  if (tid < 32) {
    float r = sbuf[tid];
#pragma unroll
    for (int off = 16; off > 0; off >>= 1) r += __shfl_xor(r, off, 32);
    if (tid == 0) sbuf[0] = r;
  }
  __syncthreads();
  float r = sbuf[0];
  __syncthreads();   // sbuf is reused by the caller
  return r;
}

// One workgroup per batch. Entire point cloud lives in LDS (96 KB of the
// 320 KB/WGP), staged by the Tensor Data Mover; per-thread dist lives in VGPRs.
__global__ __launch_bounds__(FPS_T)
void fps_kernel(const float* __restrict__ points, int* __restrict__ idx_out) {
  __shared__ float P[3 * NPTS];     // SoA: X[N], Y[N], Z[N]  (matches [3][N] source)
  __shared__ float sD[32];
  __shared__ int   sN[32];
  __shared__ int   sFar;

  const int b    = blockIdx.x;
  const int tid  = threadIdx.x;
  const int lane = tid & 31;
  const int wid  = tid >> 5;
  const float* gsrc = points + (size_t)b * 3 * NPTS;

#ifdef USE_TDM
  if (tid < 32) {                       // wave 0 only: TDM is a per-wave DMA op
    uint32_t lds = (uint32_t)(uintptr_t)(void*)P;     // low 32 bits = LDS offset
    uint64_t ga  = (uint64_t)(uintptr_t)gsrc;
    // D# group0: count=1 | lds_addr | global_addr[56:0] | type=2 ("image")
    u32x4 g0 = { 1u, lds, (uint32_t)ga, (uint32_t)(ga >> 32) | 0x80000000u };
    // D# group1: data_size=4B; tensor_dim0=N; tensor_dim1=3; tile_dim0=N;
    //            tile_dim1=3; tensor_dim0_stride=N (2D tensor, full-row tile)
    i32x8 g1 = { 0x00020000,
                 (int)((unsigned)NPTS << 16),   // tensor_dim0[15:0] @ bits 63:48
                 (int)(3u << 16),               // tensor_dim1 @ bits 95:80
                 (int)((unsigned)NPTS << 16),   // tile_dim0 @ bits 127:112
                 3,                             // tile_dim1 @ bits 143:128
                 NPTS,                          // tensor_dim0_stride[31:0]
                 0, 0 };
    i32x4 z4 = { 0, 0, 0, 0 };                  // groups 2/3 unused (2D tensor)
    i32x8 z8 = { 0, 0, 0, 0, 0, 0, 0, 0 };
    __builtin_amdgcn_tensor_load_to_lds(g0, g1, z4, z4, z8, 0);
    __builtin_amdgcn_s_wait_tensorcnt(0);
  }
#else
  for (int i = tid; i < 3 * NPTS; i += FPS_T) P[i] = gsrc[i];
#endif
  __syncthreads();

  const float* Xs = P;
  const float* Ys = P + NPTS;
  const float* Zs = P + 2 * NPTS;

  // Register-resident copies of this thread's 8 points + running min-dist.
  float px[PPT], py[PPT], pz[PPT], dist[PPT];
  float sx = 0.f, sy = 0.f, sz = 0.f;
#pragma unroll
  for (int j = 0; j < PPT; ++j) {
    int n = j * FPS_T + tid;
    px[j] = Xs[n]; py[j] = Ys[n]; pz[j] = Zs[n];
    sx += px[j]; sy += py[j]; sz += pz[j];
  }

  // Seed: farthest point from the centroid.
  float cx = blockSum(sx, sD, tid) * (1.0f / NPTS);
  float cy = blockSum(sy, sD, tid) * (1.0f / NPTS);
  float cz = blockSum(sz, sD, tid) * (1.0f / NPTS);

  float bd = -1.0f; int bn = 0;
#pragma unroll
  for (int j = 0; j < PPT; ++j) {
    int n = j * FPS_T + tid;
    float dx = px[j] - cx, dy = py[j] - cy, dz = pz[j] - cz;
    float d0 = dx * dx + dy * dy + dz * dz;
    dist[j] = 1e10f;
    if (d0 > bd || (d0 == bd && n < bn)) { bd = d0; bn = n; }
  }
  warpArgMax(bd, bn);
  if (lane == 0) { sD[wid] = bd; sN[wid] = bn; }
  __syncthreads();
  if (tid < 32) {
    float d2 = sD[tid]; int n2 = sN[tid];
    warpArgMax(d2, n2);
    if (tid == 0) sFar = n2;
  }
  __syncthreads();
  int far = sFar;

  // Sequential FPS: record far, update dist vs pts[far], argmax -> next far.
  for (int m = 0; m < MOUT; ++m) {
    if (tid == 0) idx_out[(size_t)b * MOUT + m] = far;
    float qx = Xs[far], qy = Ys[far], qz = Zs[far];   // same-address LDS broadcast
    float bd2 = -1.0f; int bn2 = 0;
#pragma unroll
    for (int j = 0; j < PPT; ++j) {
      int n = j * FPS_T + tid;
      float dx = px[j] - qx, dy = py[j] - qy, dz = pz[j] - qz;
      float d  = dx * dx + dy * dy + dz * dz;
      float dj = dist[j] < d ? dist[j] : d;
      dist[j] = dj;
      if (dj > bd2 || (dj == bd2 && n < bn2)) { bd2 = dj; bn2 = n; }
    }
    warpArgMax(bd2, bn2);
    if (lane == 0) { sD[wid] = bd2; sN[wid] = bn2; }
    __syncthreads();
    if (tid < 32) {
      float d2 = sD[tid]; int n2 = sN[tid];
      warpArgMax(d2, n2);
      if (tid == 0) sFar = n2;
    }
    __syncthreads();
    far = sFar;
  }
}

// Transpose-gather: features [B,C,N] sampled at idx -> [B,M,C], plus points
// [B,3,N] -> [B,M,3]. 32m x 64c LDS tile makes both sides coalesced.
__global__ __launch_bounds__(256)
void gather_kernel(const float* __restrict__ points,
                   const float* __restrict__ feats,
                   const int*   __restrict__ idx,
                   float* __restrict__ out_pts,
                   float* __restrict__ out_feats) {
  __shared__ int   nIdx[32];
  __shared__ float tile[64][33];   // +1 pad: conflict-free transposed reads

  const int b   = blockIdx.z;
  const int m0  = blockIdx.x * 32;
  const int c0  = blockIdx.y * 64;
  const int tid = threadIdx.x;
  const int tx  = tid & 31;        // m within tile
  const int ty  = tid >> 5;        // 0..7

  if (tid < 32) nIdx[tid] = idx[(size_t)b * MOUT + m0 + tid];
  __syncthreads();

  const int n = nIdx[tx];
#pragma unroll
  for (int k = 0; k < 8; ++k) {
    int cc = ty + k * 8;
    tile[cc][tx] = feats[((size_t)b * NCH + (size_t)(c0 + cc)) * NPTS + n];
  }
  __syncthreads();

#pragma unroll
  for (int k = 0; k < 8; ++k) {
    int e  = k * 256 + tid;
    int mm = e >> 6;
    int cw = e & 63;
    out_feats[((size_t)b * MOUT + (size_t)(m0 + mm)) * NCH + c0 + cw] = tile[cw][mm];
  }

  if (blockIdx.y == 0 && tid < 96) {
    int comp = tid >> 5;           // 0,1,2 -> x,y,z
    int mm   = tid & 31;
    out_pts[((size_t)b * MOUT + (size_t)(m0 + mm)) * 3 + comp] =
        points[((size_t)b * 3 + comp) * NPTS + nIdx[mm]];
  }
}

extern "C" void kernel_launch(void* const* d_in, const int* in_sizes, int n_in,
                              void* d_out, int out_size, void* d_ws, size_t ws_size,
                              hipStream_t stream) {
  (void)in_sizes; (void)n_in; (void)out_size; (void)ws_size;
  const float* points   = (const float*)d_in[0];   // [B,3,N] f32
  const float* features = (const float*)d_in[1];   // [B,C,N] f32
  // d_in[2] = num_points scalar (2048; fixed by the reference constants)

  int*   idx_ws    = (int*)d_ws;                          // B*M ints = 128 KB
  float* out_pts   = (float*)d_out;                       // [B,M,3]
  float* out_feats = out_pts + (size_t)BATCH * MOUT * 3;  // [B,M,C]

  fps_kernel<<<dim3(BATCH), dim3(FPS_T), 0, stream>>>(points, idx_ws);

  dim3 g(MOUT / 32, NCH / 64, BATCH);
  gather_kernel<<<g, dim3(256), 0, stream>>>(points, features, idx_ws,
                                             out_pts, out_feats);
}